// Mean_Aggregator_35845797052746
// MI455X (gfx1250) — compile-verified
//
#include <hip/hip_runtime.h>

typedef float v2f __attribute__((ext_vector_type(2)));
typedef float v8f __attribute__((ext_vector_type(8)));

#define DIM   128
#define OUTD  256
#define LDSP  144   // pair-row stride in float2 units: 288 floats = 32 banks mod 64
                    // -> lane halves (pair-rows p, p+1) hit disjoint 32-bank windows

// ---------------------------------------------------------------------------
// Kernel 1: CSR row pointers from sorted COO rows (lower_bound per row)
// ---------------------------------------------------------------------------
__global__ void rowptr_kernel(const int* __restrict__ rows, int* __restrict__ rowptr,
                              int n_rows, int n_edges) {
  int r = blockIdx.x * blockDim.x + threadIdx.x;
  if (r > n_rows) return;
  int lo = 0, hi = n_edges;
  while (lo < hi) {
    int mid = (lo + hi) >> 1;
    if (rows[mid] < r) lo = mid + 1; else hi = mid;
  }
  rowptr[r] = lo;
}

// ---------------------------------------------------------------------------
// Kernel 2: SpMM segment-sum. One wave32 per row, each lane owns 4 floats.
// Gather table (25.6 MB) lives in L2; edge stream is linear from HBM.
// ---------------------------------------------------------------------------
__global__ __launch_bounds__(256) void spmm_kernel(
    const float* __restrict__ neigh_vecs, const float* __restrict__ vals,
    const int* __restrict__ cols, const int* __restrict__ rowptr,
    float* __restrict__ means, int n_rows) {
  int wid  = (int)((blockIdx.x * blockDim.x + threadIdx.x) >> 5);
  int lane = (int)(threadIdx.x & 31);
  if (wid >= n_rows) return;

  int e0 = rowptr[wid];
  int e1 = rowptr[wid + 1];
  float4 acc = make_float4(0.f, 0.f, 0.f, 0.f);

  int e = e0;
  for (; e + 2 <= e1; e += 2) {               // 2-edge unroll for MLP
    float v0 = vals[e];
    float v1 = vals[e + 1];
    long  c0 = cols[e];
    long  c1 = cols[e + 1];
    float4 x0 = *(const float4*)(neigh_vecs + c0 * DIM + lane * 4);
    float4 x1 = *(const float4*)(neigh_vecs + c1 * DIM + lane * 4);
    acc.x = fmaf(v0, x0.x, acc.x); acc.y = fmaf(v0, x0.y, acc.y);
    acc.z = fmaf(v0, x0.z, acc.z); acc.w = fmaf(v0, x0.w, acc.w);
    acc.x = fmaf(v1, x1.x, acc.x); acc.y = fmaf(v1, x1.y, acc.y);
    acc.z = fmaf(v1, x1.z, acc.z); acc.w = fmaf(v1, x1.w, acc.w);
  }
  if (e < e1) {
    float v0 = vals[e];
    long  c0 = cols[e];
    float4 x0 = *(const float4*)(neigh_vecs + c0 * DIM + lane * 4);
    acc.x = fmaf(v0, x0.x, acc.x); acc.y = fmaf(v0, x0.y, acc.y);
    acc.z = fmaf(v0, x0.z, acc.z); acc.w = fmaf(v0, x0.w, acc.w);
  }
  *(float4*)(means + (size_t)wid * DIM + lane * 4) = acc;
}

// ---------------------------------------------------------------------------
// Kernel 3: out[:, cb:cb+128] = relu(X @ W) via V_WMMA_F32_16X16X4_F32.
// W is staged in LDS as pair-rows: lw2[p][c] = {W[2p][c], W[2p+1][c]} so each
// B fragment (rows k,k+1 of one 16-col tile) is ONE aligned ds_load_b64 into
// an even VGPR pair -- no repacking movs between WMMAs.
// blockIdx.y: 0 -> (self_vecs, self_weights, cols 0..127)
//             1 -> (neigh_means, neigh_weights, cols 128..255)
// Block = 256 threads = 8 waves; each wave owns one 16-row strip (16x128 C).
// ---------------------------------------------------------------------------
__global__ __launch_bounds__(256) void gemm_relu_kernel(
    const float* __restrict__ X0, const float* __restrict__ W0,
    const float* __restrict__ X1, const float* __restrict__ W1,
    float* __restrict__ out, int n_rows) {
  extern __shared__ float lw[];   // [64 pair-rows][LDSP float2] = 73728 B

  const float* X = (blockIdx.y == 0) ? X0 : X1;
  const float* W = (blockIdx.y == 0) ? W0 : W1;
  const int col_base = (int)blockIdx.y * DIM;

  // Stage W: read row-major float4 (coalesced), scatter into pair-row layout.
  for (int i = threadIdx.x; i < (DIM * DIM) / 4; i += 256) {
    int idx = i * 4;
    int r = idx >> 7;          // W row
    int c = idx & (DIM - 1);   // W col (multiple of 4)
    float4 w4 = *(const float4*)(W + idx);
    float* dst = lw + ((r >> 1) * LDSP + c) * 2 + (r & 1);
    dst[0] = w4.x; dst[2] = w4.y; dst[4] = w4.z; dst[6] = w4.w;
  }
  __syncthreads();

  const int wave = (int)(threadIdx.x >> 5);
  const int lane = (int)(threadIdx.x & 31);
  const int m0   = (int)blockIdx.x * 128 + wave * 16;
  if (m0 >= n_rows) return;          // wave-uniform: EXEC stays all-1s for WMMA

  const int hi    = lane >> 4;       // lane half: K pair (k0,k0+1) vs (k0+2,k0+3)
  int       mlane = m0 + (lane & 15);
  if (mlane >= n_rows) mlane = n_rows - 1;   // clamp loads; stores guarded below
  const float* xrow = X + (size_t)mlane * DIM;

  v8f acc[8];
#pragma unroll
  for (int n = 0; n < 8; ++n) acc[n] = 0.0f;

  for (int k0 = 0; k0 < DIM; k0 += 4) {
    // A fragment: lanes 0-15 -> (K=k0,K=k0+1), lanes 16-31 -> (K=k0+2,K=k0+3)
    float2 av = *(const float2*)(xrow + k0 + 2 * hi);
    v2f a; a.x = av.x; a.y = av.y;

    // B fragments: pair-row (k0/2 + hi), 16-col tile n, one b64 load each.
    const float* wb = lw + (((k0 >> 1) + hi) * LDSP + (lane & 15)) * 2;
#pragma unroll
    for (int n = 0; n < 8; ++n) {
      v2f b = *(const v2f*)(wb + n * 32);
      acc[n] = __builtin_amdgcn_wmma_f32_16x16x4_f32(
          false, a, false, b, (short)0, acc[n], false, false);
    }
  }

  // Epilogue: C layout -> VGPR v holds M = m0 + v + 8*hi, N = n*16 + (lane&15)
  const int rbase = m0 + 8 * hi;
  const int cbase = col_base + (lane & 15);
#pragma unroll
  for (int n = 0; n < 8; ++n) {
#pragma unroll
    for (int v = 0; v < 8; ++v) {
      int row = rbase + v;
      if (row < n_rows) {
        float val = acc[n][v];
        out[(size_t)row * OUTD + cbase + n * 16] = val > 0.f ? val : 0.f;
      }
    }
  }
}

// ---------------------------------------------------------------------------
extern "C" void kernel_launch(void* const* d_in, const int* in_sizes, int n_in,
                              void* d_out, int out_size, void* d_ws, size_t ws_size,
                              hipStream_t stream) {
  const float* neigh_vecs = (const float*)d_in[0];
  const float* self_vecs  = (const float*)d_in[1];
  const float* adj_vals   = (const float*)d_in[2];
  const int*   adj_rows   = (const int*)d_in[3];
  const int*   adj_cols   = (const int*)d_in[4];
  const float* neigh_w    = (const float*)d_in[5];
  const float* self_w     = (const float*)d_in[6];
  float* out = (float*)d_out;

  const int n_nodes = in_sizes[0] / DIM;
  const int n_edges = in_sizes[2];

  // workspace: [rowptr: (n+1) ints][align 512][neigh_means: n*128 f32]
  int* rowptr = (int*)d_ws;
  size_t off = (((size_t)(n_nodes + 1) * sizeof(int)) + 511) & ~(size_t)511;
  float* means = (float*)((char*)d_ws + off);

  rowptr_kernel<<<(n_nodes + 256) / 256, 256, 0, stream>>>(
      adj_rows, rowptr, n_nodes, n_edges);

  spmm_kernel<<<(n_nodes * 32 + 255) / 256, 256, 0, stream>>>(
      neigh_vecs, adj_vals, adj_cols, rowptr, means, n_nodes);

  dim3 grid((n_nodes + 127) / 128, 2);
  gemm_relu_kernel<<<grid, 256, 64 * LDSP * 2 * sizeof(float), stream>>>(
      self_vecs, self_w, means, neigh_w, out, n_nodes);
}